// CRF2oDependency_5514738008073
// MI455X (gfx1250) — compile-verified
//
#include <hip/hip_runtime.h>
#include <float.h>

// CRF2o dependency inside algorithm for MI455X (gfx1250).
// One workgroup per batch element; the three NxN DP tables live in the
// 320KB-per-WGP LDS (194KB) plus a 64KB double-buffered TDM staging area for
// sibling-score stripes. Tensor Data Mover (tensor_load_to_lds) prefetches
// width w+1's strided sib rows while width w computes; sync via TENSORcnt.
// Wave32 lane-parallel logsumexp with __shfl_xor butterflies.

#define NTOK   128
#define PITCH  129            // +1 pad: column walks rotate across 64 LDS banks
#define NEGV   (-1.0e9f)
#define NWAVES 16
#define BLOCK  (NWAVES * 32)
#define TBL    (NTOK * PITCH) // floats per DP table
#define STAGEH 4096           // floats per stripe (max w*(128-w) = 4096)
#define STAGEF (2 * STAGEH)   // floats per staging buffer (il + ir)
#define SIBSTRIDE 16513       // element stride between consecutive m rows

#if defined(__has_builtin)
# if __has_builtin(__builtin_amdgcn_tensor_load_to_lds) && \
     __has_builtin(__builtin_amdgcn_s_wait_tensorcnt)
#  define USE_TDM 1
# endif
#endif
#ifndef USE_TDM
# define USE_TDM 0
#endif

typedef unsigned int u32x4 __attribute__((ext_vector_type(4)));
typedef int          i32x8 __attribute__((ext_vector_type(8)));
typedef int          i32x4 __attribute__((ext_vector_type(4)));

__device__ __forceinline__ unsigned ldsOffsetOf(const void* p) {
    // generic -> AS(3): the integer value of an LDS pointer is the DS-space
    // byte offset, which is exactly what the TDM D# lds_addr field wants.
    return (unsigned)(unsigned long long)(const __attribute__((address_space(3))) void*)p;
}

#if USE_TDM
// Issue one 2D TDM load: tileH rows of tileW floats, row stride SIBSTRIDE
// elements, packed contiguously into LDS at ldsByteOff.
__device__ __forceinline__ void tdm_load_stripe(const float* gptr, unsigned ldsByteOff,
                                                int tileW, int tileH) {
    unsigned long long ga = (unsigned long long)(size_t)gptr;
    const unsigned TD0 = 1u << 24;     // tensor_dim0 (huge: stripes are in-bounds)
    u32x4 g0;
    g0.x = 1u;                                           // count=1 (valid user D#)
    g0.y = ldsByteOff;                                   // lds_addr (bytes)
    g0.z = (unsigned)(ga & 0xFFFFFFFFull);               // global_addr[31:0]
    g0.w = (unsigned)((ga >> 32) & 0x01FFFFFFull)        // global_addr[56:32]
         | (2u << 30);                                   // type=2 ("image")
    i32x8 g1;
    g1[0] = 0x00020000;                                  // data_size=2 -> 4 bytes
    g1[1] = (int)((TD0 & 0xFFFFu) << 16);                // tensor_dim0 lo16
    g1[2] = (int)(((TD0 >> 16) & 0xFFFFu)                // tensor_dim0 hi16
         | (((unsigned)tileH & 0xFFFFu) << 16));         // tensor_dim1 lo16
    g1[3] = (int)(((unsigned)tileW & 0xFFFFu) << 16);    // tensor_dim1 hi16=0, tile_dim0
    g1[4] = (int)((unsigned)tileH & 0xFFFFu);            // tile_dim1; tile_dim2=0 (2D)
    g1[5] = (int)SIBSTRIDE;                              // tensor_dim0_stride lo32
    g1[6] = 0;                                           // stride hi16; dim1_stride=0
    g1[7] = 0;
    i32x4 gz4 = {0, 0, 0, 0};                            // groups 2/3 unused (2D tile)
    i32x8 gz8 = {0, 0, 0, 0, 0, 0, 0, 0};                // trailing group (clang-23 form)
    // amdgpu-toolchain (clang-23 / therock-10.0) 6-arg form:
    // (u32x4 g0, i32x8 g1, i32x4, i32x4, i32x8, i32 cpol)
    __builtin_amdgcn_tensor_load_to_lds(g0, g1, gz4, gz4, gz8, 0);
}
#endif

__device__ __forceinline__ void lse_acc(float& M, float& S, float x) {
    float Mn = fmaxf(M, x);
    S = S * __expf(M - Mn) + __expf(x - Mn);
    M = Mn;
}

__device__ __forceinline__ float wave_lse(float M, float S) {
#pragma unroll
    for (int off = 16; off > 0; off >>= 1) {
        float M2 = __shfl_xor(M, off, 32);
        float S2 = __shfl_xor(S, off, 32);
        float Mn = fmaxf(M, M2);
        S = S * __expf(M - Mn) + S2 * __expf(M2 - Mn);
        M = Mn;
    }
    return M + __logf(S);
}

__global__ void crf2o_init_acc(float* acc) {
    acc[0] = 0.0f;  // sum over batch of (logZ_b - score_b)
    acc[1] = 0.0f;  // mask.sum()
}

__global__ __launch_bounds__(BLOCK)
void crf2o_inside_kernel(const float* __restrict__ s_arc,
                         const float* __restrict__ s_sib,
                         const unsigned char* __restrict__ mask,
                         const int* __restrict__ arcs,
                         const int* __restrict__ sibs,
                         float* __restrict__ acc) {
    const int b    = blockIdx.x;
    const int tid  = threadIdx.x;
    const int wave = tid >> 5;
    const int lane = tid & 31;

    extern __shared__ float lds[];
    float* sI    = lds;             // incomplete spans
    float* sS    = lds + TBL;       // sibling spans
    float* sC    = lds + 2 * TBL;   // complete spans
    float* stage = lds + 3 * TBL;   // TDM staging: [2][STAGEF]
    __shared__ float scoreSh;
    __shared__ int   lenSh;

    const float* arcB = s_arc + (size_t)b * NTOK * NTOK;
    const float* sibB = s_sib + (size_t)b * NTOK * NTOK * NTOK;

    if (tid == 0) { scoreSh = 0.0f; lenSh = 0; }

    // ---- init DP tables ----
    for (int idx = tid; idx < TBL; idx += BLOCK) {
        sI[idx] = NEGV; sS[idx] = NEGV; sC[idx] = NEGV;
    }
    __syncthreads();
    for (int i = tid; i < NTOK; i += BLOCK) sC[i * PITCH + i] = 0.0f;

    // ---- gold score + sentence length for this batch element ----
    float scLocal = 0.0f;
    int   lenLocal = 0;
    for (int i = tid; i < NTOK; i += BLOCK) {
        int mk = mask[b * NTOK + i];
        lenLocal += mk;
        if (mk) {
            int a = arcs[b * NTOK + i];
            scLocal += arcB[i * NTOK + a];
            int s = sibs[b * NTOK + i];
            if (s > 0) scLocal += sibB[((size_t)i * NTOK + a) * NTOK + s];
        }
    }
    if (scLocal != 0.0f) atomicAdd(&scoreSh, scLocal);
    if (lenLocal)        atomicAdd(&lenSh, lenLocal);
    __syncthreads();
    const int len = lenSh;

    // ---- inside recursion over span widths ----
    for (int w = 1; w < NTOK; ++w) {
        const int n = NTOK - w;

#if USE_TDM
        // Kick off DMA of width w+1's sibling stripes into the alternate
        // staging buffer; buffer was last read in phase 1 of w-1 (sealed by
        // barrier). DMA overlaps all of width w's compute.
        if (wave == 0 && w + 1 < NTOK) {
            const int u  = w + 1;
            const int nu = NTOK - u;
            float* bufN = stage + ((u & 1) * STAGEF);
            if (u - 1 > 0)
                tdm_load_stripe(sibB + 128 * u + 1, ldsOffsetOf(bufN), u - 1, nu);
            tdm_load_stripe(sibB + (size_t)16384 * u, ldsOffsetOf(bufN + STAGEH), u, nu);
        }
        const float* stIl = stage + ((w & 1) * STAGEF); // rows of (w-1) floats
        const float* stIr = stIl + STAGEH;              // rows of w floats
#endif

        // PHASE 1: left/right incomplete spans + sibling spans (write diag +/-w
        // of sI and sS; all reads touch diagonals |k| < w, so no intra-phase race)
        for (int m = wave; m < n; m += NWAVES) {
#if !USE_TDM
            const float* sib_il = sibB + ((size_t)m * NTOK + (m + w)) * NTOK;
            const float* sib_ir = sibB + ((size_t)(m + w) * NTOK + m) * NTOK;
            if (w + 1 < NTOK && m < NTOK - (w + 1)) {
                const char* p1 = (const char*)(sibB + ((size_t)m * NTOK + (m + w + 1)) * NTOK + (1 + m));
                const char* p2 = (const char*)(sibB + ((size_t)(m + w + 1) * NTOK + m) * NTOK + m);
                for (int off = lane * 128; off < (w + 1) * 4; off += 32 * 128) {
                    __builtin_prefetch(p1 + off, 0, 1);
                    __builtin_prefetch(p2 + off, 0, 1);
                }
            }
#endif
            // il: I(j<-i) candidates; last column replaced by C+C seed
            float Mi = -FLT_MAX, Si = 0.0f;
            for (int j = lane; j < w; j += 32) {
                float t;
                if (j == w - 1) {
                    t = (m == 0) ? 0.0f
                                 : (sC[(w + m) * PITCH + (w + m)] + sC[m * PITCH + (w - 1 + m)]);
                } else {
#if USE_TDM
                    float sib = stIl[m * (w - 1) + j];
#else
                    float sib = sib_il[1 + m + j];
#endif
                    t = sI[(w + m) * PITCH + (1 + m + j)] + sS[(1 + m + j) * PITCH + m] + sib;
                }
                lse_acc(Mi, Si, t);
            }
            float il = wave_lse(Mi, Si);

            // ir: I(i->j); column 0 replaced by C+C seed; row m==0 masked
            float Mr = -FLT_MAX, Sr = 0.0f;
            for (int j = lane; j < w; j += 32) {
                float t;
                if (j == 0)      t = sC[m * PITCH + m] + sC[(m + w) * PITCH + (m + 1)];
                else if (m == 0) t = NEGV;
                else {
#if USE_TDM
                    float sib = stIr[m * w + j];
#else
                    float sib = sib_ir[m + j];
#endif
                    t = sI[m * PITCH + (m + j)] + sS[(m + j) * PITCH + (m + w)] + sib;
                }
                lse_acc(Mr, Sr, t);
            }
            float ir = wave_lse(Mr, Sr);

            // slr: sibling span
            float Ms = -FLT_MAX, Ss = 0.0f;
            for (int j = lane; j < w; j += 32) {
                lse_acc(Ms, Ss, sC[m * PITCH + (m + j)] + sC[(w + m) * PITCH + (1 + m + j)]);
            }
            float sl = wave_lse(Ms, Ss);

            if (lane == 0) {
                sI[(m + w) * PITCH + m] = il + arcB[m * NTOK + (m + w)];
                sI[m * PITCH + (m + w)] = ir + arcB[(m + w) * NTOK + m];
                sS[(m + w) * PITCH + m] = sl;
                sS[m * PITCH + (m + w)] = sl;
            }
        }
        __syncthreads();

        // PHASE 2: complete spans (read fresh sI diag +/-w, write sC diag +/-w)
        for (int m = wave; m < n; m += NWAVES) {
            float Ml = -FLT_MAX, Sl = 0.0f;
            float Mr2 = -FLT_MAX, Sr2 = 0.0f;
            for (int j = lane; j < w; j += 32) {
                lse_acc(Ml,  Sl,  sC[(m + j) * PITCH + m]     + sI[(m + w) * PITCH + (m + j)]);
                lse_acc(Mr2, Sr2, sI[m * PITCH + (1 + m + j)] + sC[(1 + m + j) * PITCH + (w + m)]);
            }
            float cl = wave_lse(Ml, Sl);
            float cr = wave_lse(Mr2, Sr2);
            if (lane == 0) {
                sC[(m + w) * PITCH + m] = cl;
                // fold root-length restriction into the m==0 write of diag +w
                sC[m * PITCH + (m + w)] = (m == 0 && len != w) ? NEGV : cr;
            }
        }
#if USE_TDM
        if (wave == 0) __builtin_amdgcn_s_wait_tensorcnt(0);  // w+1 stripes landed
#endif
        __syncthreads();
    }

    if (tid == 0) {
        float partial = sC[0 * PITCH + len] - scoreSh;  // logZ_b - score_b
        atomicAdd(&acc[0], partial);
        atomicAdd(&acc[1], (float)len);
    }
}

__global__ void crf2o_finalize(const float* acc, float* out) {
    out[0] = acc[0] / acc[1];
}

extern "C" void kernel_launch(void* const* d_in, const int* in_sizes, int n_in,
                              void* d_out, int out_size, void* d_ws, size_t ws_size,
                              hipStream_t stream) {
    (void)in_sizes; (void)n_in; (void)out_size; (void)ws_size;
    const float*         s_arc = (const float*)d_in[0];
    const float*         s_sib = (const float*)d_in[1];
    const unsigned char* mask  = (const unsigned char*)d_in[2];  // jax bool = 1 byte
    const int*           arcs  = (const int*)d_in[3];
    const int*           sibs  = (const int*)d_in[4];
    float* out = (float*)d_out;
    float* acc = (float*)d_ws;

    const int B = 32;
    const size_t ldsBytes = (size_t)(3 * TBL + 2 * STAGEF) * sizeof(float); // 263680 B

    // allow >64KB dynamic LDS (CDNA5 WGP has 320KB); host-side attr, capture-safe
    (void)hipFuncSetAttribute((const void*)crf2o_inside_kernel,
                              hipFuncAttributeMaxDynamicSharedMemorySize, (int)ldsBytes);

    crf2o_init_acc<<<1, 1, 0, stream>>>(acc);
    crf2o_inside_kernel<<<B, BLOCK, ldsBytes, stream>>>(s_arc, s_sib, mask, arcs, sibs, acc);
    crf2o_finalize<<<1, 1, 0, stream>>>(acc, out);
}